// MattingCNN_16707422781578
// MI455X (gfx1250) — compile-verified
//
#include <hip/hip_runtime.h>
#include <math.h>

// ---------------------------------------------------------------------------
// Matting CG solver for MI455X (gfx1250, wave32).
// Sparse operators kept factored; scatter via hardware fp32 global atomics.
// Dot products use V_WMMA_F32_16X16X4_F32 (A = x*y as 16x4, B = ones).
// ---------------------------------------------------------------------------

typedef __attribute__((ext_vector_type(2))) float v2f;
typedef __attribute__((ext_vector_type(8))) float v8f;

__device__ __forceinline__ void atomAddF(float* p, float v) {
    __hip_atomic_fetch_add(p, v, __ATOMIC_RELAXED, __HIP_MEMORY_SCOPE_AGENT);
}

// -------------------------- setup kernels ----------------------------------

// diag = KU*kToUconf + lmbda*known ; r = p = b = diag*kToU ; x = 0 ; rs_cm = 0
__global__ void k_setup_vec(const float* KU, const float* kToUconf,
                            const float* lmbda, const float* known,
                            const float* kToU,
                            float* diag, float* r, float* p, float* x,
                            float* rs_cm, float* s, int n) {
    int i = blockIdx.x * blockDim.x + threadIdx.x;
    if (i < n) {
        float dk = KU[i] * kToUconf[i] + lmbda[0] * known[i];
        float b  = dk * kToU[i];
        diag[i] = dk;
        r[i] = b;
        p[i] = b;
        x[i] = 0.0f;
        rs_cm[i] = 0.0f;
    }
    if (i == 0) { s[0] = 0.0f; s[1] = 0.0f; s[2] = 0.0f; s[3] = 0.0f; }
}

// cm_vals = CM_weights[row]*Wcm_data ; rs_cm[row] += cm_vals
__global__ void k_setup_cm(const float* CMw, const float* Wdat,
                           const int* row, float* cm_vals, float* rs_cm,
                           int nnz) {
    int e = blockIdx.x * blockDim.x + threadIdx.x;
    if (e >= nnz) return;
    int ro = row[e];
    float cv = CMw[ro] * Wdat[e];
    cm_vals[e] = cv;
    atomAddF(&rs_cm[ro], cv);
}

// diag[neigh_i] += 0.5*w*sum_j (F[j][i]+F[i][j])   (rs_m folded into diag)
__global__ void k_setup_loc(const float* flows, const float* locw,
                            const int* inInd, float* diag,
                            int nloc, int width) {
    int k = blockIdx.x * blockDim.x + threadIdx.x;
    if (k >= nloc) return;
    int c = inInd[k];
    float w = locw[c];
    const int offs[9] = {-1 - width, -1, -1 + width, -width, 0, width,
                         1 - width, 1, 1 + width};
    float F[9][9];
#pragma unroll
    for (int j = 0; j < 9; ++j)
#pragma unroll
        for (int i = 0; i < 9; ++i)
            F[j][i] = flows[(j * 9 + i) * nloc + k];
#pragma unroll
    for (int i = 0; i < 9; ++i) {
        float rs = 0.0f;
#pragma unroll
        for (int j = 0; j < 9; ++j) rs += F[j][i] + F[i][j];
        atomAddF(&diag[c + offs[i]], 0.5f * w * rs);
    }
}

// diag[in] += sum_q 0.5*f ; diag[nb] += 0.5*f   (rs_c folded into diag)
__global__ void k_setup_iu(const float* IUfl, const float* iuw,
                           const int* inInd, const int* neighInd,
                           float* diag, int nloc) {
    int k = blockIdx.x * blockDim.x + threadIdx.x;
    if (k >= nloc) return;
    int a = inInd[k];
    float w = iuw[a];
    float acc = 0.0f;
#pragma unroll
    for (int q = 0; q < 5; ++q) {
        float f = 0.5f * IUfl[k * 5 + q] * w;
        acc += f;
        atomAddF(&diag[neighInd[k * 5 + q]], f);
    }
    atomAddF(&diag[a], acc);
}

// -------------------------- per-CG-step kernels ----------------------------

// Ap = diag*p ; Lv = rs_cm*p ; rotate scalars: s0 <- s2, s1 = s2 = 0
__global__ void k_mv_init(const float* diag, const float* rs_cm,
                          const float* p, float* Ap, float* Lv,
                          float* s, int n) {
    int i = blockIdx.x * blockDim.x + threadIdx.x;
    if (i < n) {
        float pv = p[i];
        Ap[i] = diag[i] * pv;
        Lv[i] = rs_cm[i] * pv;
    }
    if (i == 0) { s[0] = s[2]; s[1] = 0.0f; s[2] = 0.0f; }
}

// Lv[row] -= cm_vals * p[col]
__global__ void k_cm_pass1(const float* cm_vals, const int* row, const int* col,
                           const float* p, float* Lv, int nnz) {
    int e = blockIdx.x * blockDim.x + threadIdx.x;
    if (e >= nnz) return;
    atomAddF(&Lv[row[e]], -cm_vals[e] * p[col[e]]);
}

// Ap += rs_cm * Lv   (elementwise, runs after pass1 completes in-stream)
__global__ void k_cm_elem(const float* rs_cm, const float* Lv, float* Ap, int n) {
    int i = blockIdx.x * blockDim.x + threadIdx.x;
    if (i < n) Ap[i] += rs_cm[i] * Lv[i];
}

// Ap[col] -= cm_vals * Lv[row]
__global__ void k_cm_pass2(const float* cm_vals, const int* row, const int* col,
                           const float* Lv, float* Ap, int nnz) {
    int e = blockIdx.x * blockDim.x + threadIdx.x;
    if (e >= nnz) return;
    atomAddF(&Ap[col[e]], -cm_vals[e] * Lv[row[e]]);
}

// Ap[neigh_i] -= 0.5*w * sum_j (F[j][i]+F[i][j]) * v[neigh_j]
__global__ void k_loc_mv(const float* flows, const float* locw,
                         const int* inInd, const float* v, float* Ap,
                         int nloc, int width) {
    int k = blockIdx.x * blockDim.x + threadIdx.x;
    if (k >= nloc) return;
    int c = inInd[k];
    float w = locw[c];
    const int offs[9] = {-1 - width, -1, -1 + width, -width, 0, width,
                         1 - width, 1, 1 + width};
    int ni[9];
    float vg[9];
#pragma unroll
    for (int i = 0; i < 9; ++i) { ni[i] = c + offs[i]; vg[i] = v[ni[i]]; }
    float F[9][9];
#pragma unroll
    for (int j = 0; j < 9; ++j)
#pragma unroll
        for (int i = 0; i < 9; ++i)
            F[j][i] = flows[(j * 9 + i) * nloc + k];
#pragma unroll
    for (int i = 0; i < 9; ++i) {
        float y = 0.0f;
#pragma unroll
        for (int j = 0; j < 9; ++j) y += (F[j][i] + F[i][j]) * vg[j];
        atomAddF(&Ap[ni[i]], -0.5f * w * y);
    }
}

// Ap[in] -= sum_q f*v[nb] ; Ap[nb] -= f*v[in]   with f = 0.5*IUfl*w
__global__ void k_iu_mv(const float* IUfl, const float* iuw,
                        const int* inInd, const int* neighInd,
                        const float* v, float* Ap, int nloc) {
    int k = blockIdx.x * blockDim.x + threadIdx.x;
    if (k >= nloc) return;
    int a = inInd[k];
    float w = iuw[a];
    float va = v[a];
    float acc = 0.0f;
#pragma unroll
    for (int q = 0; q < 5; ++q) {
        int b = neighInd[k * 5 + q];
        float f = 0.5f * IUfl[k * 5 + q] * w;
        acc += f * v[b];
        atomAddF(&Ap[b], -f * va);
    }
    atomAddF(&Ap[a], -acc);
}

// x += alpha*p ; r -= alpha*Ap   with alpha = s[0]/s[1]
__global__ void k_axpy(const float* s, const float* p, const float* Ap,
                       float* x, float* r, int n) {
    int i = blockIdx.x * blockDim.x + threadIdx.x;
    if (i >= n) return;
    float alpha = s[0] / s[1];
    x[i] += alpha * p[i];
    r[i] -= alpha * Ap[i];
}

// p = r + beta*p   with beta = s[2]/s[0]
__global__ void k_pupd(const float* s, const float* r, float* p, int n) {
    int i = blockIdx.x * blockDim.x + threadIdx.x;
    if (i >= n) return;
    float beta = s[2] / s[0];
    p[i] = r[i] + beta * p[i];
}

// ---------------------- WMMA dot-product reduction -------------------------
// Per wave, per iteration: z = x*y over a 64-element chunk is packed as the
// 16x4 f32 A operand (lane L<16: M=L, K=0/1 in the 2 VGPRs; lane L>=16:
// M=L-16, K=2/3). B = all-ones (layout independent). Then
// C[m][n] = sum_k z[4m+k], identical across n, and C accumulates across
// chunks. Lane<16 VGPR v holds S_v, lane>=16 holds S_{v+8}; so
// total = lanesum(lane 0) + lanesum(lane 16).
__global__ void k_dot_wmma(const float* x, const float* y, float* out, int n) {
    const int lane = threadIdx.x & 31;
    const int wavesPerBlock = blockDim.x >> 5;
    const int waveId = blockIdx.x * wavesPerBlock + (threadIdx.x >> 5);
    const int totalWaves = gridDim.x * wavesPerBlock;
    const int chunks = n >> 6;               // 64 elements per chunk
    const int m = lane & 15;
    const int half = lane >> 4;

    v8f c = {};
    v2f ones; ones.x = 1.0f; ones.y = 1.0f;

    for (int ch = waveId; ch < chunks; ch += totalWaves) {   // uniform per wave
        int e0 = (ch << 6) + 4 * m + 2 * half;               // 8B aligned
        float2 xv = *(const float2*)(x + e0);
        float2 yv = *(const float2*)(y + e0);
        v2f a; a.x = xv.x * yv.x; a.y = xv.y * yv.y;
        c = __builtin_amdgcn_wmma_f32_16x16x4_f32(
                /*neg_a=*/false, a, /*neg_b=*/false, ones,
                /*c_mod=*/(short)0, c, /*reuse_a=*/false, /*reuse_b=*/false);
    }

    float local = c[0] + c[1] + c[2] + c[3] + c[4] + c[5] + c[6] + c[7];
    float tot = __shfl(local, 0, 32) + __shfl(local, 16, 32);
    if (lane == 0) atomAddF(out, tot);

    // scalar tail for n % 64 (none for this problem size, kept for safety)
    int tail0 = chunks << 6;
    for (int i = tail0 + blockIdx.x * blockDim.x + threadIdx.x; i < n;
         i += gridDim.x * blockDim.x)
        atomAddF(out, x[i] * y[i]);
}

// ---------------------------------------------------------------------------

extern "C" void kernel_launch(void* const* d_in, const int* in_sizes, int n_in,
                              void* d_out, int out_size, void* d_ws, size_t ws_size,
                              hipStream_t stream) {
    const float* CM_weights  = (const float*)d_in[0];
    const float* LOC_weights = (const float*)d_in[1];
    const float* IU_weights  = (const float*)d_in[2];
    const float* KU_weights  = (const float*)d_in[3];
    const float* lmbda       = (const float*)d_in[4];
    const float* kToUconf    = (const float*)d_in[5];
    const float* known       = (const float*)d_in[6];
    const float* kToU        = (const float*)d_in[7];
    const float* Wcm_data    = (const float*)d_in[8];
    const float* LOC_flows   = (const float*)d_in[9];
    const float* IU_flows    = (const float*)d_in[10];
    const int*   Wcm_row     = (const int*)d_in[11];
    const int*   Wcm_col     = (const int*)d_in[12];
    const int*   LOC_inInd   = (const int*)d_in[13];
    const int*   IU_inInd    = (const int*)d_in[14];
    const int*   IU_neighInd = (const int*)d_in[15];

    const int N    = in_sizes[0];            // 147456
    const int NNZ  = in_sizes[8];            // N_LOC * 20
    const int NLOC = in_sizes[13];           // 73728
    const int width = (int)(sqrt((double)N) + 0.5);   // 384 (H==W grid)
    const int CG_STEPS = 30;                 // matches reference setup

    // workspace layout (floats)
    float* ws     = (float*)d_ws;
    float* cm_vals = ws;                     // NNZ
    float* rs_cm   = cm_vals + NNZ;          // N
    float* diag    = rs_cm + N;              // N  (diag_ku + rs_m + rs_c)
    float* r       = diag + N;               // N
    float* p       = r + N;                  // N
    float* Ap      = p + N;                  // N
    float* Lv      = Ap + N;                 // N
    float* s       = Lv + N;                 // 4 scalars: rs_old, pAp, rs_new
    float* x       = (float*)d_out;          // solution accumulates in d_out

    const int T = 256;
    const int gN   = (N + T - 1) / T;
    const int gNNZ = (NNZ + T - 1) / T;
    const int gLOC = (NLOC + T - 1) / T;
    const int gDOT = 32;                     // 256 waves total

    // ---- setup ----
    k_setup_vec<<<gN, T, 0, stream>>>(KU_weights, kToUconf, lmbda, known, kToU,
                                      diag, r, p, x, rs_cm, s, N);
    k_setup_cm<<<gNNZ, T, 0, stream>>>(CM_weights, Wcm_data, Wcm_row,
                                       cm_vals, rs_cm, NNZ);
    k_setup_loc<<<gLOC, T, 0, stream>>>(LOC_flows, LOC_weights, LOC_inInd,
                                        diag, NLOC, width);
    k_setup_iu<<<gLOC, T, 0, stream>>>(IU_flows, IU_weights, IU_inInd,
                                       IU_neighInd, diag, NLOC);
    // s[2] = dot(b,b); first k_mv_init rotates it into s[0]
    k_dot_wmma<<<gDOT, T, 0, stream>>>(r, r, &s[2], N);

    // ---- CG iterations ----
    for (int it = 0; it < CG_STEPS; ++it) {
        k_mv_init<<<gN, T, 0, stream>>>(diag, rs_cm, p, Ap, Lv, s, N);
        k_cm_pass1<<<gNNZ, T, 0, stream>>>(cm_vals, Wcm_row, Wcm_col, p, Lv, NNZ);
        k_cm_elem<<<gN, T, 0, stream>>>(rs_cm, Lv, Ap, N);
        k_cm_pass2<<<gNNZ, T, 0, stream>>>(cm_vals, Wcm_row, Wcm_col, Lv, Ap, NNZ);
        k_loc_mv<<<gLOC, T, 0, stream>>>(LOC_flows, LOC_weights, LOC_inInd,
                                         p, Ap, NLOC, width);
        k_iu_mv<<<gLOC, T, 0, stream>>>(IU_flows, IU_weights, IU_inInd,
                                        IU_neighInd, p, Ap, NLOC);
        k_dot_wmma<<<gDOT, T, 0, stream>>>(p, Ap, &s[1], N);   // pAp
        k_axpy<<<gN, T, 0, stream>>>(s, p, Ap, x, r, N);
        k_dot_wmma<<<gDOT, T, 0, stream>>>(r, r, &s[2], N);    // rs_new
        k_pupd<<<gN, T, 0, stream>>>(s, r, p, N);
    }
}